// GM_GAT_13572096655878
// MI455X (gfx1250) — compile-verified
//
#include <hip/hip_runtime.h>
#include <hip/hip_bf16.h>

#define HEADS 3
#define HID   64
#define FDIM  192   // HEADS*HID
#define INDIM 256
#define NCLS  40
#define NEG_SLOPE 0.2f

typedef __attribute__((ext_vector_type(2))) float v2f;
typedef __attribute__((ext_vector_type(8))) float v8f;

// ---------- ordered-uint encoding for float atomic max ----------
__device__ __forceinline__ unsigned enc_f32(float f) {
  unsigned u = __float_as_uint(f);
  return (u & 0x80000000u) ? ~u : (u | 0x80000000u);
}
__device__ __forceinline__ float dec_f32(unsigned u) {
  return (u & 0x80000000u) ? __uint_as_float(u & 0x7FFFFFFFu)
                           : __uint_as_float(~u);
}
#define ENC_NEG_INF 0x007FFFFFu   // enc_f32(-inf)

__device__ __forceinline__ float leaky(float x) {
  return x > 0.f ? x : NEG_SLOPE * x;
}

// =====================================================================
// WMMA fp32 GEMM:  C[M x NC] = A[M x K] * B[K x NC] (+ bias)
// One wave per 16x48 output strip (3 column tiles -> 3 independent
// accumulator chains per A fragment; branch-free tail via clamp+select).
// M must be a multiple of 16 (50000 = 3125*16). Fragment layout per ISA
// 7.12.2: A/B element K = reg + 2*(lane>=16); A row / B col = lane&15;
// C/D row M = reg + 8*(lane>=16), col = lane&15.
// =====================================================================
#define NT_PER_WAVE 3
__global__ void gat_gemm_wmma_f32(const float* __restrict__ A,
                                  const float* __restrict__ B,
                                  const float* __restrict__ bias,
                                  float* __restrict__ C,
                                  int M, int K, int NC, int ngroups) {
  int wave = (int)((blockIdx.x * blockDim.x + threadIdx.x) >> 5);
  int mtiles = M >> 4;
  if (wave >= mtiles * ngroups) return;           // uniform per wave
  int mt = wave / ngroups;
  int g  = wave % ngroups;
  int lane = (int)(threadIdx.x & 31);
  int half = lane >> 4;
  int rc   = lane & 15;

  int arow = mt * 16 + rc;                        // A row for this lane
  int col[NT_PER_WAVE];
  int colc[NT_PER_WAVE];
  bool colOK[NT_PER_WAVE];
#pragma unroll
  for (int j = 0; j < NT_PER_WAVE; ++j) {
    col[j]   = g * (16 * NT_PER_WAVE) + j * 16 + rc;
    colOK[j] = (col[j] < NC);
    colc[j]  = colOK[j] ? col[j] : (NC - 1);      // clamp: in-bounds load
  }

  const float* Ap = A + (size_t)arow * K;
  v8f acc[NT_PER_WAVE];
#pragma unroll
  for (int j = 0; j < NT_PER_WAVE; ++j) acc[j] = (v8f){};

  for (int k = 0; k < K; k += 4) {
    int ka = k + 2 * half;
    v2f a;
    a.x = Ap[ka];
    a.y = Ap[ka + 1];
    const float* B0 = B + (size_t)ka * NC;
    const float* B1 = B0 + NC;
#pragma unroll
    for (int j = 0; j < NT_PER_WAVE; ++j) {
      float bx = B0[colc[j]];
      float by = B1[colc[j]];
      v2f b;
      b.x = colOK[j] ? bx : 0.f;                  // v_cndmask, no branch
      b.y = colOK[j] ? by : 0.f;
      acc[j] = __builtin_amdgcn_wmma_f32_16x16x4_f32(
          false, a, false, b, (short)0, acc[j], false, false);
    }
  }

#pragma unroll
  for (int j = 0; j < NT_PER_WAVE; ++j) {
    if (!colOK[j]) continue;
    float bv = bias ? bias[col[j]] : 0.f;
#pragma unroll
    for (int r = 0; r < 8; ++r) {
      int orow = mt * 16 + r + 8 * half;
      C[(size_t)orow * NC + col[j]] = acc[j][r] + bv;
    }
  }
}

// ---------------- per-node attention scores -------------------------
__global__ void gat_attn_scores(const float* __restrict__ H,
                                const float* __restrict__ att_s,
                                const float* __restrict__ att_d,
                                float* __restrict__ asrc,
                                float* __restrict__ adst, int N3) {
  int idx = blockIdx.x * blockDim.x + threadIdx.x;
  if (idx >= N3) return;
  int n = idx / HEADS, h = idx % HEADS;
  const float* hp = H + (size_t)n * FDIM + h * HID;
  const float* sp = att_s + h * HID;
  const float* dp = att_d + h * HID;
  float s0 = 0.f, s1 = 0.f;
  for (int c = 0; c < HID; ++c) {
    float v = hp[c];
    s0 += v * sp[c];
    s1 += v * dp[c];
  }
  asrc[idx] = s0;
  adst[idx] = s1;
}

// ---------------- init AGG / softmax stats --------------------------
__global__ void gat_zero_f32(float* __restrict__ p, long long n) {
  long long i = (long long)blockIdx.x * blockDim.x + threadIdx.x;
  if (i < n) p[i] = 0.f;
}
__global__ void gat_init_stats(unsigned* __restrict__ mx,
                               float* __restrict__ sm, int n) {
  int i = blockIdx.x * blockDim.x + threadIdx.x;
  if (i < n) { mx[i] = ENC_NEG_INF; sm[i] = 0.f; }
}

// ---------------- edge passes ---------------------------------------
__device__ __forceinline__ void edge_sd(const int* ei, int E, int e,
                                        int& s, int& d) {
  if (e < E) { s = ei[e]; d = ei[E + e]; }
  else       { s = d = e - E; }           // appended self-loops
}

__global__ void gat_edge_max(const int* __restrict__ ei, int E, int Etot,
                             const float* __restrict__ asrc,
                             const float* __restrict__ adst,
                             unsigned* __restrict__ mx) {
  int e = blockIdx.x * blockDim.x + threadIdx.x;
  if (e >= Etot) return;
  int s, d; edge_sd(ei, E, e, s, d);
  for (int h = 0; h < HEADS; ++h) {
    float l = leaky(asrc[s * HEADS + h] + adst[d * HEADS + h]);
    atomicMax(&mx[d * HEADS + h], enc_f32(l));
  }
}

__global__ void gat_edge_expsum(const int* __restrict__ ei, int E, int Etot,
                                const float* __restrict__ asrc,
                                const float* __restrict__ adst,
                                const unsigned* __restrict__ mx,
                                float* __restrict__ ev,
                                float* __restrict__ sm) {
  int e = blockIdx.x * blockDim.x + threadIdx.x;
  if (e >= Etot) return;
  int s, d; edge_sd(ei, E, e, s, d);
  for (int h = 0; h < HEADS; ++h) {
    float l = leaky(asrc[s * HEADS + h] + adst[d * HEADS + h]);
    float x = __expf(l - dec_f32(mx[d * HEADS + h]));
    ev[(size_t)e * HEADS + h] = x;
    atomicAdd(&sm[d * HEADS + h], x);
  }
}

// one thread per (edge, channel): AGG[dst][c] += H[src][c] * alpha(e, c/64)
__global__ void gat_edge_agg(const int* __restrict__ ei, int E,
                             long long total,
                             const float* __restrict__ H,
                             const float* __restrict__ ev,
                             const float* __restrict__ sm,
                             float* __restrict__ AGG) {
  long long idx = (long long)blockIdx.x * blockDim.x + threadIdx.x;
  if (idx >= total) return;
  int e = (int)(idx / FDIM);
  int c = (int)(idx % FDIM);
  int s, d; edge_sd(ei, E, e, s, d);
  int h = c >> 6;
  float coef = ev[(size_t)e * HEADS + h] / sm[d * HEADS + h];
  atomicAdd(&AGG[(size_t)d * FDIM + c], H[(size_t)s * FDIM + c] * coef);
}

// ---------------- bias + relu ---------------------------------------
__global__ void gat_bias_relu(float* __restrict__ p,
                              const float* __restrict__ b, long long total) {
  long long i = (long long)blockIdx.x * blockDim.x + threadIdx.x;
  if (i >= total) return;
  int c = (int)(i % FDIM);
  float v = p[i] + b[c];
  p[i] = v > 0.f ? v : 0.f;
}

// =====================================================================
extern "C" void kernel_launch(void* const* d_in, const int* in_sizes, int n_in,
                              void* d_out, int out_size, void* d_ws,
                              size_t ws_size, hipStream_t stream) {
  const float* x    = (const float*)d_in[0];
  const int*   ei   = (const int*)  d_in[1];
  const float* W1   = (const float*)d_in[2];
  const float* as1  = (const float*)d_in[3];
  const float* ad1  = (const float*)d_in[4];
  const float* b1   = (const float*)d_in[5];
  const float* W2   = (const float*)d_in[6];
  const float* as2  = (const float*)d_in[7];
  const float* ad2  = (const float*)d_in[8];
  const float* b2   = (const float*)d_in[9];
  const float* Wout = (const float*)d_in[10];
  const float* bout = (const float*)d_in[11];

  const int N    = in_sizes[0] / INDIM;   // 50000
  const int E    = in_sizes[1] / 2;       // 800000
  const int Etot = E + N;                 // with self-loops

  // ---- workspace carve-up (floats) ----
  float*    H    = (float*)d_ws;                    // N*FDIM
  float*    AGG  = H   + (size_t)N * FDIM;          // N*FDIM
  float*    asrc = AGG + (size_t)N * FDIM;          // N*HEADS
  float*    adst = asrc + (size_t)N * HEADS;        // N*HEADS
  unsigned* mx   = (unsigned*)(adst + (size_t)N * HEADS); // N*HEADS
  float*    sm   = (float*)mx + (size_t)N * HEADS;  // N*HEADS
  float*    ev   = sm + (size_t)N * HEADS;          // Etot*HEADS

  const int TB = 256;
  const int n3 = N * HEADS;
  const long long nf  = (long long)N * FDIM;
  const long long etf = (long long)Etot * FDIM;
  const int waves_per_blk = 4;            // 128 threads

  auto gemm = [&](const float* A, const float* B, const float* bias, float* C,
                  int K, int NC) {
    int ntiles  = (NC + 15) / 16;
    int ngroups = (ntiles + NT_PER_WAVE - 1) / NT_PER_WAVE;
    int waves   = (N / 16) * ngroups;
    int blocks  = (waves + waves_per_blk - 1) / waves_per_blk;
    hipLaunchKernelGGL(gat_gemm_wmma_f32, dim3(blocks),
                       dim3(waves_per_blk * 32), 0, stream,
                       A, B, bias, C, N, K, NC, ngroups);
  };

  auto gat_layer = [&](const float* X, int K, const float* W,
                       const float* att_s, const float* att_d,
                       const float* bias) {
    gemm(X, W, nullptr, H, K, FDIM);
    hipLaunchKernelGGL(gat_zero_f32, dim3((unsigned)((nf + TB - 1) / TB)),
                       dim3(TB), 0, stream, AGG, nf);
    hipLaunchKernelGGL(gat_init_stats, dim3((n3 + TB - 1) / TB), dim3(TB), 0,
                       stream, mx, sm, n3);
    hipLaunchKernelGGL(gat_attn_scores, dim3((n3 + TB - 1) / TB), dim3(TB), 0,
                       stream, H, att_s, att_d, asrc, adst, n3);
    hipLaunchKernelGGL(gat_edge_max, dim3((Etot + TB - 1) / TB), dim3(TB), 0,
                       stream, ei, E, Etot, asrc, adst, mx);
    hipLaunchKernelGGL(gat_edge_expsum, dim3((Etot + TB - 1) / TB), dim3(TB),
                       0, stream, ei, E, Etot, asrc, adst, mx, ev, sm);
    hipLaunchKernelGGL(gat_edge_agg, dim3((unsigned)((etf + TB - 1) / TB)),
                       dim3(TB), 0, stream, ei, E, etf, H, ev, sm, AGG);
    hipLaunchKernelGGL(gat_bias_relu, dim3((unsigned)((nf + TB - 1) / TB)),
                       dim3(TB), 0, stream, AGG, bias, nf);
  };

  // layer 1: X = input x (K = 256); result (post relu) lands in AGG
  gat_layer(x, INDIM, W1, as1, ad1, b1);
  // layer 2: X = AGG (K = 192); gemm reads AGG before it is re-zeroed
  gat_layer(AGG, FDIM, W2, as2, ad2, b2);
  // readout: d_out = AGG @ Wout + bout   (N x 40)
  gemm(AGG, Wout, bout, (float*)d_out, FDIM, NCLS);
}